// RGCNEncoder_30623116821147
// MI455X (gfx1250) — compile-verified
//
#include <hip/hip_runtime.h>

typedef __attribute__((ext_vector_type(2))) float v2f;
typedef __attribute__((ext_vector_type(8))) float v8f;

#define THREADS 256

// ---------------------------------------------------------------------------
// Y[nrows,64] = X[nrows,64] @ W[64,64], fp32, via V_WMMA_F32_16X16X4_F32.
// One wave32 per 16-row tile; 4 accumulators cover the 64 output columns.
// A fragment layout (ISA 7.12.2, 32-bit A 16x4): lane m = lane%16,
//   VGPR0 = A[m][2*(lane/16)], VGPR1 = A[m][2*(lane/16)+1].
// B fragment (4x16, row striped across lanes): lane n = lane%16,
//   VGPR v, half h -> K = v + 2h.
// C/D (16x16 f32, 8 VGPRs): lane n = lane%16, VGPR v -> row v + 8*(lane/16).
// ---------------------------------------------------------------------------
__global__ __launch_bounds__(THREADS) void rgcn_gemm_wmma(
    const float* __restrict__ X, const float* __restrict__ W,
    float* __restrict__ Y, int nrows)
{
    const int wave  = threadIdx.x >> 5;
    const int lane  = threadIdx.x & 31;
    const int mtile = blockIdx.x * (THREADS / 32) + wave;
    if (mtile * 16 >= nrows) return;                // wave-uniform guard

    const int m     = lane & 15;                    // A row / B,C col within tile
    const int khalf = lane >> 4;                    // 0 or 1

    const float* Arow = X + (size_t)(mtile * 16 + m) * 64;

    v8f acc0 = {}, acc1 = {}, acc2 = {}, acc3 = {};
#pragma unroll
    for (int kk = 0; kk < 16; ++kk) {
        const int kbase = kk * 4 + khalf * 2;
        v2f a = *(const v2f*)(Arow + kbase);        // 64-bit load, reused 4x
        const float* W0 = W + (size_t)kbase * 64;
        v2f b0, b1, b2, b3;
        b0.x = W0[m +  0];  b0.y = W0[64 + m +  0];
        b1.x = W0[m + 16];  b1.y = W0[64 + m + 16];
        b2.x = W0[m + 32];  b2.y = W0[64 + m + 32];
        b3.x = W0[m + 48];  b3.y = W0[64 + m + 48];
        acc0 = __builtin_amdgcn_wmma_f32_16x16x4_f32(false, a, false, b0, (short)0, acc0, false, false);
        acc1 = __builtin_amdgcn_wmma_f32_16x16x4_f32(false, a, false, b1, (short)0, acc1, false, false);
        acc2 = __builtin_amdgcn_wmma_f32_16x16x4_f32(false, a, false, b2, (short)0, acc2, false, false);
        acc3 = __builtin_amdgcn_wmma_f32_16x16x4_f32(false, a, false, b3, (short)0, acc3, false, false);
    }

#pragma unroll
    for (int v = 0; v < 8; ++v) {
        const int row = mtile * 16 + v + 8 * khalf;
        float* Yr = Y + (size_t)row * 64 + m;
        Yr[ 0] = acc0[v];
        Yr[16] = acc1[v];
        Yr[32] = acc2[v];
        Yr[48] = acc3[v];
    }
}

// Zero the per-relation accumulators (sums[N*64] and cnt[N]).
__global__ __launch_bounds__(THREADS) void rgcn_zero(
    float* __restrict__ sums, float* __restrict__ cnt, int n)
{
    int idx = blockIdx.x * blockDim.x + threadIdx.x;
    if (idx < n * 64) sums[idx] = 0.0f;
    if (idx < n)      cnt[idx]  = 0.0f;
}

// Scatter: 4 threads per edge, 16 floats each. Hardware f32 atomics.
__global__ __launch_bounds__(THREADS) void rgcn_edge_scatter(
    const int* __restrict__ src, const int* __restrict__ dst,
    const int* __restrict__ etype, int rel, int ne,
    const float* __restrict__ xr,
    float* __restrict__ sums, float* __restrict__ cnt)
{
    int idx = blockIdx.x * blockDim.x + threadIdx.x;
    int e = idx >> 2;
    if (e >= ne) return;
    if (etype[e] != rel) return;
    int c = (idx & 3) * 16;
    int s = src[e];
    int d = dst[e];
    const float4* g = (const float4*)(xr + (size_t)s * 64 + c);
    float* out = sums + (size_t)d * 64 + c;
#pragma unroll
    for (int q = 0; q < 4; ++q) {
        float4 v = g[q];
        unsafeAtomicAdd(out + q * 4 + 0, v.x);
        unsafeAtomicAdd(out + q * 4 + 1, v.y);
        unsafeAtomicAdd(out + q * 4 + 2, v.z);
        unsafeAtomicAdd(out + q * 4 + 3, v.w);
    }
    if ((idx & 3) == 0) unsafeAtomicAdd(cnt + d, 1.0f);
}

// hidden += cnt>0 ? sums/cnt + bias : 0   (cnt>0 implies cnt>=1)
__global__ __launch_bounds__(THREADS) void rgcn_update(
    float* __restrict__ hid, const float* __restrict__ sums,
    const float* __restrict__ cnt, const float* __restrict__ bias, int n)
{
    int idx = blockIdx.x * blockDim.x + threadIdx.x;
    if (idx >= n * 64) return;
    int node = idx >> 6;
    int col  = idx & 63;
    float c = cnt[node];
    if (c > 0.0f) hid[idx] += sums[idx] / c + bias[col];
}

__global__ __launch_bounds__(THREADS) void rgcn_copy(
    float* __restrict__ out, const float* __restrict__ in, int n)
{
    int idx = blockIdx.x * blockDim.x + threadIdx.x;
    if (idx < n) out[idx] = in[idx];
}

__global__ __launch_bounds__(THREADS) void rgcn_relu(
    float* __restrict__ out, const float* __restrict__ in, int n)
{
    int idx = blockIdx.x * blockDim.x + threadIdx.x;
    if (idx < n) out[idx] = fmaxf(in[idx], 0.0f);
}

extern "C" void kernel_launch(void* const* d_in, const int* in_sizes, int n_in,
                              void* d_out, int out_size, void* d_ws, size_t ws_size,
                              hipStream_t stream)
{
    const int*   edge_index = (const int*)d_in[0];    // [2, NE]
    const int*   edge_type  = (const int*)d_in[1];    // [NE]
    const float* embeddings = (const float*)d_in[2];  // [N, 64]
    const float* weights    = (const float*)d_in[3];  // [L, 16, 64, 64]
    const float* biases     = (const float*)d_in[4];  // [L, 16, 64]

    const int R = 8, E = 64;
    const int NE = in_sizes[1];
    const int N  = in_sizes[2] / E;
    const int L  = in_sizes[4] / (2 * R * E);

    // Workspace layout (fp32): xr | sums | cnt | hidden0 | x1
    float* f       = (float*)d_ws;
    float* xr      = f;  f += (size_t)N * E;
    float* sums    = f;  f += (size_t)N * E;
    float* cnt     = f;  f += ((size_t)N + 63) & ~63ull;
    float* hidden0 = f;  f += (size_t)N * E;
    float* x1      = f;

    const int nelem = N * E;
    dim3 blk(THREADS);
    dim3 gElem((nelem + THREADS - 1) / THREADS);
    dim3 gEdge((NE * 4 + THREADS - 1) / THREADS);
    int mtiles = (N + 15) / 16;
    dim3 gGemm((mtiles + (THREADS / 32) - 1) / (THREADS / 32));

    for (int l = 0; l < L; ++l) {
        const float* xin = (l == 0) ? embeddings : x1;
        float* hid = (l == L - 1) ? (float*)d_out : hidden0;

        rgcn_copy<<<gElem, blk, 0, stream>>>(hid, xin, nelem);

        for (int inv = 0; inv < 2; ++inv) {
            const int* s_idx = edge_index + (size_t)(1 - inv) * NE; // sources
            const int* d_idx = edge_index + (size_t)inv * NE;       // destinations
            for (int r = 0; r < R; ++r) {
                const int widx = l * 2 * R + (r + R * inv);
                rgcn_zero<<<gElem, blk, 0, stream>>>(sums, cnt, N);
                rgcn_gemm_wmma<<<gGemm, blk, 0, stream>>>(
                    xin, weights + (size_t)widx * E * E, xr, N);
                rgcn_edge_scatter<<<gEdge, blk, 0, stream>>>(
                    s_idx, d_idx, edge_type, r, NE, xr, sums, cnt);
                rgcn_update<<<gElem, blk, 0, stream>>>(
                    hid, sums, cnt, biases + (size_t)widx * E, N);
            }
        }

        if (l < L - 1)
            rgcn_relu<<<gElem, blk, 0, stream>>>(x1, hid, nelem);
    }
}